// ATEM_26465588478182
// MI455X (gfx1250) — compile-verified
//
#include <hip/hip_runtime.h>
#include <math.h>

// ---------------------------------------------------------------------------
// MI455X plan:
//   Stage 1 (bandwidth-bound, ~100 MB @ 23.3 TB/s ≈ 4.3 us floor):
//     per-wave fp32 WMMA (V_WMMA_F32_16X16X4_F32) Gram accumulation
//     D[16x16] += A(16x4) * B(4x16), A = features, B = w * features.
//   Stage 2: 256 scalar 9x9 Jacobi eigensolves (trivial work).
// fp32 WMMA is chosen deliberately: smallest-eigenvector extraction is
// numerically sensitive; bf16/fp8 matrix paths would corrupt the nullspace.
// ---------------------------------------------------------------------------

typedef __attribute__((ext_vector_type(2))) float v2f;
typedef __attribute__((ext_vector_type(8))) float v8f;

#define BATCH   256
#define NPTS    20000
#define GCHUNK  8                   // workgroups per batch
#define WAVES   8                   // waves per workgroup (256 threads, wave32)
#define KT      64                  // points per wave-tile
#define KSTR    66                  // padded LDS row stride (floats, even for b64)
#define WAVE_LDS (16 * KSTR + 68)   // 16 feature rows + weight row = 1124 floats
#define RED_OFF  (WAVES * WAVE_LDS) // 8992 floats
#define LDS_FLOATS (RED_OFF + WAVES * 256) // + 8 x 16x16 reduce slots = 11040

__global__ __launch_bounds__(256)
void atem_accum(const float* __restrict__ x_in,
                const float* __restrict__ wgt,
                float* __restrict__ partial)
{
    __shared__ float lds[LDS_FLOATS];

    const int tid  = threadIdx.x;
    const int wave = tid >> 5;
    const int lane = tid & 31;
    const int b    = blockIdx.x / GCHUNK;
    const int c    = blockIdx.x % GCHUNK;
    const int cpts = NPTS / GCHUNK;          // 2500
    const int beg  = c * cpts;
    const int end  = beg + cpts;

    const float4* xb = reinterpret_cast<const float4*>(x_in) + (size_t)b * NPTS;
    const float*  wb = wgt + (size_t)b * NPTS;

    // Zero per-wave LDS once: feature rows 9..15 must stay zero so the
    // pad lanes (m >= 9) feed exact zeros into the WMMA.
    for (int i = tid; i < RED_OFF; i += 256) lds[i] = 0.0f;
    __syncthreads();

    float* Aw = lds + wave * WAVE_LDS;       // 16 x KSTR, feature-major
    float* Ww = Aw + 16 * KSTR;              // 64(+pad) weights

    v8f acc = {0.f, 0.f, 0.f, 0.f, 0.f, 0.f, 0.f, 0.f};

    const int m  = lane & 15;                // feature index this lane supplies
    const int hi = lane >> 4;                // 0: K=0,1   1: K=2,3
    const float* Arow = Aw + m * KSTR;

    const int niter = (cpts + WAVES * KT - 1) / (WAVES * KT);   // 5, uniform
    for (int it = 0; it < niter; ++it) {
        const int base = beg + (it * WAVES + wave) * KT;

        // ---- phase A: coalesced global loads -> features in LDS ----
        #pragma unroll
        for (int half = 0; half < 2; ++half) {
            const int j  = base + half * 32 + lane;   // point index
            const int kk = half * 32 + lane;          // slot in tile
            float4 q; q.x = q.y = q.z = q.w = 0.f;
            float  w = 0.f;
            if (j < end) { q = xb[j]; w = wb[j]; }
            const float x1 = q.x, y1 = q.y, x2 = q.z, y2 = q.w;
            Aw[0 * KSTR + kk] = x2 * x1;
            Aw[1 * KSTR + kk] = x2 * y1;
            Aw[2 * KSTR + kk] = x2;
            Aw[3 * KSTR + kk] = y2 * x1;
            Aw[4 * KSTR + kk] = y2 * y1;
            Aw[5 * KSTR + kk] = y2;
            Aw[6 * KSTR + kk] = x1;
            Aw[7 * KSTR + kk] = y1;
            Aw[8 * KSTR + kk] = 1.0f;   // w=0 for invalid points kills the term
            Ww[kk] = w;
        }
        __syncthreads();

        // ---- phase B: fp32 WMMA, 4 points per instruction ----
        // B = diag(w) * A and the B lane index equals the A lane index,
        // so B operand = (w0*a.x, w1*a.y): no second staging pass.
        #pragma unroll
        for (int g = 0; g < KT / 4; ++g) {
            const int kb = g * 4 + hi * 2;
            v2f a  = *reinterpret_cast<const v2f*>(Arow + kb);
            v2f w2 = *reinterpret_cast<const v2f*>(Ww + kb);
            v2f bm; bm.x = w2.x * a.x; bm.y = w2.y * a.y;
            acc = __builtin_amdgcn_wmma_f32_16x16x4_f32(
                      false, a, false, bm, (short)0, acc, false, false);
        }
        __syncthreads();
    }

    // ---- deterministic in-block reduction of the 8 wave accumulators ----
    float* red = lds + RED_OFF;
    #pragma unroll
    for (int r = 0; r < 8; ++r) {
        const int M = r + hi * 8;            // documented 16x16 f32 C/D layout
        red[wave * 256 + M * 16 + m] = acc[r];
    }
    __syncthreads();
    if (tid < 32) {
        for (int e = lane; e < 256; e += 32) {
            float s = 0.f;
            #pragma unroll
            for (int wv = 0; wv < WAVES; ++wv) s += red[wv * 256 + e];
            partial[(size_t)blockIdx.x * 256 + e] = s;
        }
    }
}

// ---------------------------------------------------------------------------
// Stage 2: per-batch 9x9 cyclic Jacobi eigensolve, min-eigenvalue vector.
// ---------------------------------------------------------------------------
__global__ __launch_bounds__(64)
void atem_eig(const float* __restrict__ partial, float* __restrict__ out)
{
    const int b = blockIdx.x * blockDim.x + threadIdx.x;
    if (b >= BATCH) return;

    float A[9][9];
    float V[9][9];
    for (int i = 0; i < 9; ++i)
        for (int j = 0; j < 9; ++j) {
            float s = 0.f;
            for (int cc = 0; cc < GCHUNK; ++cc)
                s += partial[((size_t)(b * GCHUNK + cc)) * 256 + i * 16 + j];
            A[i][j] = s;
            V[i][j] = (i == j) ? 1.f : 0.f;
        }

    for (int sweep = 0; sweep < 12; ++sweep) {
        for (int p = 0; p < 8; ++p)
            for (int q = p + 1; q < 9; ++q) {
                const float apq = A[p][q];
                if (fabsf(apq) < 1e-28f) continue;
                const float tau = (A[q][q] - A[p][p]) / (2.0f * apq);
                const float t   = (tau >= 0.f ? 1.f : -1.f) /
                                  (fabsf(tau) + sqrtf(tau * tau + 1.f));
                const float cth = rsqrtf(t * t + 1.f);
                const float sth = t * cth;
                for (int k = 0; k < 9; ++k) {        // A = A * G
                    const float akp = A[k][p], akq = A[k][q];
                    A[k][p] = cth * akp - sth * akq;
                    A[k][q] = sth * akp + cth * akq;
                }
                for (int k = 0; k < 9; ++k) {        // A = G^T * A
                    const float apk = A[p][k], aqk = A[q][k];
                    A[p][k] = cth * apk - sth * aqk;
                    A[q][k] = sth * apk + cth * aqk;
                }
                for (int k = 0; k < 9; ++k) {        // V = V * G
                    const float vkp = V[k][p], vkq = V[k][q];
                    V[k][p] = cth * vkp - sth * vkq;
                    V[k][q] = sth * vkp + cth * vkq;
                }
            }
    }

    int jm = 0; float dm = A[0][0];
    for (int j = 1; j < 9; ++j) { const float d = A[j][j]; if (d < dm) { dm = d; jm = j; } }
    float nrm = 0.f;
    for (int i = 0; i < 9; ++i) nrm += V[i][jm] * V[i][jm];
    const float inv = rsqrtf(fmaxf(nrm, 1e-30f));
    for (int i = 0; i < 9; ++i) out[b * 9 + i] = V[i][jm] * inv;
}

// ---------------------------------------------------------------------------
extern "C" void kernel_launch(void* const* d_in, const int* in_sizes, int n_in,
                              void* d_out, int out_size, void* d_ws, size_t ws_size,
                              hipStream_t stream)
{
    const float* x_in = (const float*)d_in[0];   // (256,1,20000,4) fp32
    const float* wgt  = (const float*)d_in[1];   // (256,20000)     fp32
    float* out     = (float*)d_out;              // (256,9)         fp32
    float* partial = (float*)d_ws;               // 2048 x 256 floats = 2 MB

    atem_accum<<<BATCH * GCHUNK, 256, 0, stream>>>(x_in, wgt, partial);
    atem_eig<<<(BATCH + 63) / 64, 64, 0, stream>>>(partial, out);
}